// RevePositionBankWrapper_22471268892727
// MI455X (gfx1250) — compile-verified
//
#include <hip/hip_runtime.h>
#include <stdint.h>

// Problem geometry (fixed by the reference).
#define NTOK   8192   // rows of one_hot / out
#define VOCABN 8192   // one-hot width == weight rows
#define DIMN   1024   // embedding dim
#define ROWS_PER_BLOCK 16
#define THREADS 256
#define CHUNKS (VOCABN / (THREADS * 4))   // 8 x f32x4 per thread per row

typedef float        f32x4 __attribute__((ext_vector_type(4)));
typedef unsigned int u32x4 __attribute__((ext_vector_type(4)));
typedef int          i32x8 __attribute__((ext_vector_type(8)));
typedef int          i32x4 __attribute__((ext_vector_type(4)));

#if __has_builtin(__builtin_amdgcn_tensor_load_to_lds) && \
    __has_builtin(__builtin_amdgcn_tensor_store_from_lds)
#define USE_TDM 1
#else
#define USE_TDM 0
#endif

// Build TDM D# group 0: count=1 (valid), optional gather_mode, lds byte addr,
// 57-bit global byte addr, type=2 ("image").
__device__ __forceinline__ u32x4 tdm_group0(unsigned ctrl, unsigned lds_byte_addr,
                                            unsigned long long gaddr) {
  u32x4 g;
  g.x = ctrl;                                   // [1:0]=count, [31]=gather_mode
  g.y = lds_byte_addr;                          // [63:32] lds_addr
  g.z = (unsigned)(gaddr & 0xffffffffull);      // global_addr[31:0]
  g.w = ((unsigned)((gaddr >> 32) & 0x1ffffffull)) | (2u << 30); // ga[56:32] | type=2
  return g;
}

__global__ __launch_bounds__(THREADS) void position_bank_gather_kernel(
    const float* __restrict__ one_hot,   // [NTOK, VOCABN] f32, true one-hot rows
    const float* __restrict__ weight,    // [VOCABN, DIMN]  f32
    float* __restrict__ out) {           // [NTOK, DIMN]    f32
  __shared__ unsigned s_idx[ROWS_PER_BLOCK];
  extern __shared__ float stage[];       // ROWS_PER_BLOCK * DIMN floats (64 KB)

  const int tid  = threadIdx.x;
  const int row0 = blockIdx.x * ROWS_PER_BLOCK;

  // ---- Phase 1: recover the index of the single 1.0 in each one-hot row. ----
  // All 8 b128 NT loads for a row are issued before any use (8 in flight per
  // thread), then a branchless cndmask chain folds the unique nonzero position
  // into a +1-biased code; exactly one thread per row takes the single
  // conditional ds_store (16 divergent branches total, not 128).
  for (int r = 0; r < ROWS_PER_BLOCK; ++r) {
    const f32x4* rowp =
        (const f32x4*)(one_hot + (size_t)(row0 + r) * (size_t)VOCABN);
    f32x4 v[CHUNKS];
#pragma unroll
    for (int c = 0; c < CHUNKS; ++c) {
      v[c] = __builtin_nontemporal_load(&rowp[c * THREADS + tid]);
    }
    unsigned enc = 0u;   // 0 = not found; else column+1
#pragma unroll
    for (int c = 0; c < CHUNKS; ++c) {
      const unsigned base = (unsigned)(c * THREADS * 4 + tid * 4) + 1u;
      enc = (v[c].x != 0.0f) ? (base + 0u) : enc;
      enc = (v[c].y != 0.0f) ? (base + 1u) : enc;
      enc = (v[c].z != 0.0f) ? (base + 2u) : enc;
      enc = (v[c].w != 0.0f) ? (base + 3u) : enc;
    }
    if (enc) s_idx[r] = enc - 1u;
  }
  __syncthreads();

#if USE_TDM
  // ---- Phase 2: wave 0 drives the Tensor Data Mover. ----
  if (tid < 32) {  // exactly wave 0 under wave32; other waves branch around
    const unsigned lds_off = (unsigned)(unsigned long long)(uintptr_t)stage;

    // 16-bit gather row indices into D# groups 2/3 (row_index_0..15).
    i32x4 g2, g3;
    g2.x = (int)(s_idx[0]  | (s_idx[1]  << 16));
    g2.y = (int)(s_idx[2]  | (s_idx[3]  << 16));
    g2.z = (int)(s_idx[4]  | (s_idx[5]  << 16));
    g2.w = (int)(s_idx[6]  | (s_idx[7]  << 16));
    g3.x = (int)(s_idx[8]  | (s_idx[9]  << 16));
    g3.y = (int)(s_idx[10] | (s_idx[11] << 16));
    g3.z = (int)(s_idx[12] | (s_idx[13] << 16));
    g3.w = (int)(s_idx[14] | (s_idx[15] << 16));

    // Gather-load descriptor: 16 rows of weight (DIMN f32 each) -> LDS stage.
    // ctrl: count=1, gather_mode(bit31)=1, 16-bit indices (bit30=0).
    u32x4 ld0 = tdm_group0(1u | (1u << 31), lds_off,
                           (unsigned long long)(uintptr_t)weight);
    i32x8 ld1;
    ld1[0] = 0x20000;                         // data_size=2 (4 B); mask=0; no pad
    ld1[1] = (int)((DIMN & 0xffff) << 16);    // tensor_dim0[15:0]=1024
    ld1[2] = (int)(((VOCABN & 0xffff) << 16) | (DIMN >> 16)); // dim0 hi | dim1 lo
    ld1[3] = (int)((DIMN & 0xffff) << 16);    // dim1 hi=0 | tile_dim0=1024
    ld1[4] = ROWS_PER_BLOCK;                  // tile_dim1 = #valid indices = 16
    ld1[5] = DIMN;                            // tensor_dim0_stride[31:0]=1024
    ld1[6] = 0;                               // stride0 hi | stride1 lo (ignored)
    ld1[7] = 0;

    // Store descriptor: contiguous 16*1024 f32 (64 KB) LDS -> out[row0..row0+15].
    const unsigned TOT = ROWS_PER_BLOCK * DIMN;   // 16384 elements
    u32x4 st0 = tdm_group0(
        1u, lds_off,
        (unsigned long long)(uintptr_t)(out + (size_t)row0 * (size_t)DIMN));
    i32x8 st1;
    st1[0] = 0x20000;                         // data_size = 4 B
    st1[1] = (int)((TOT & 0xffff) << 16);     // tensor_dim0[15:0]=16384
    st1[2] = (int)((1u << 16) | (TOT >> 16)); // dim0 hi=0 | tensor_dim1 lo=1
    st1[3] = (int)((TOT & 0xffff) << 16);     // dim1 hi=0 | tile_dim0=16384
    st1[4] = 1;                               // tile_dim1=1
    st1[5] = (int)TOT;                        // tensor_dim0_stride=16384
    st1[6] = 0;
    st1[7] = 0;
    i32x4 z4 = {0, 0, 0, 0};
    i32x8 z8 = {0, 0, 0, 0, 0, 0, 0, 0};      // extra operand of the 6-arg form

    __builtin_amdgcn_tensor_load_to_lds(ld0, ld1, g2, g3, z8, 0);
    __builtin_amdgcn_s_wait_tensorcnt(0);   // LDS tile resident before store
    __builtin_amdgcn_tensor_store_from_lds(st0, st1, z4, z4, z8, 0);
    __builtin_amdgcn_s_wait_tensorcnt(0);   // drain before wave exit
  }
#else
  // Fallback: plain coalesced b128 gather copy (still bandwidth-optimal).
  for (int r = 0; r < ROWS_PER_BLOCK; ++r) {
    const unsigned idx = s_idx[r];
    const f32x4* src = (const f32x4*)(weight + (size_t)idx * (size_t)DIMN);
    f32x4* dst = (f32x4*)(out + (size_t)(row0 + r) * (size_t)DIMN);
    dst[tid] = src[tid];   // 256 threads * 4 f32 = 1024 f32 per row
  }
#endif
}

extern "C" void kernel_launch(void* const* d_in, const int* in_sizes, int n_in,
                              void* d_out, int out_size, void* d_ws, size_t ws_size,
                              hipStream_t stream) {
  (void)in_sizes; (void)n_in; (void)out_size; (void)d_ws; (void)ws_size;
  const float* one_hot = (const float*)d_in[0];   // [NTOK, VOCABN] f32
  const float* weight  = (const float*)d_in[1];   // [VOCABN, DIMN] f32
  float*       out     = (float*)d_out;           // [NTOK, DIMN]   f32

  const dim3 grid(NTOK / ROWS_PER_BLOCK);         // 512 workgroups
  const dim3 block(THREADS);                      // 8 waves (wave32)
  const size_t shmem = (size_t)ROWS_PER_BLOCK * DIMN * sizeof(float); // 64 KB
  position_bank_gather_kernel<<<grid, block, shmem, stream>>>(one_hot, weight, out);
}